// ListMLELoss_40046275068653
// MI455X (gfx1250) — compile-verified
//
#include <hip/hip_runtime.h>
#include <hip/hip_bf16.h>
#include <math.h>

typedef __attribute__((ext_vector_type(2))) float v2f;
typedef __attribute__((ext_vector_type(8))) float v8f;

#define ROW_LEN 4096
#define NT      512            // 16 waves of 32
#define NWAVES  (NT / 32)

__device__ __forceinline__ unsigned int f32_sortable(float f) {
    unsigned int u = __float_as_uint(f);
    // monotone map: ascending uint  <=>  ascending float
    return (u & 0x80000000u) ? ~u : (u | 0x80000000u);
}

__global__ __launch_bounds__(NT)
void listmle_row_kernel(const float* __restrict__ preds,
                        const float* __restrict__ labels,
                        float* __restrict__ row_loss)
{
    __shared__ unsigned long long kv[ROW_LEN];   // 32 KB: sort keys+payload
    __shared__ float sv[ROW_LEN];                // 16 KB: exp values / suffix sums
    __shared__ float tt[ROW_LEN / 16];           // 1 KB : tile totals
    __shared__ float wred[NWAVES];
    __shared__ float rowmax_sh;

    const int row = blockIdx.x;
    const int tid = threadIdx.x;
    const float* p  = preds  + (size_t)row * ROW_LEN;
    const float* lb = labels + (size_t)row * ROW_LEN;

    // ---- load + pack: sort key = descending label (via ~sortable), payload = pred bits
    for (int i = tid; i < ROW_LEN; i += NT) {
        unsigned int hk = ~f32_sortable(lb[i]);
        unsigned int pv = __float_as_uint(p[i]);
        kv[i] = ((unsigned long long)hk << 32) | (unsigned long long)pv;
    }
    __syncthreads();

    // ---- bitonic sort (ascending u64) fully in LDS
    for (int k = 2; k <= ROW_LEN; k <<= 1) {
        for (int j = k >> 1; j > 0; j >>= 1) {
            for (int i = tid; i < ROW_LEN; i += NT) {
                int ixj = i ^ j;
                if (ixj > i) {
                    unsigned long long a = kv[i], b = kv[ixj];
                    bool up = ((i & k) == 0);
                    if ((a > b) == up) { kv[i] = b; kv[ixj] = a; }
                }
            }
            __syncthreads();
        }
    }

    // ---- row max of preds (permutation invariant; read from sorted payloads)
    float m = -INFINITY;
    for (int i = tid; i < ROW_LEN; i += NT)
        m = fmaxf(m, __uint_as_float((unsigned int)(kv[i] & 0xFFFFFFFFull)));
    for (int off = 16; off > 0; off >>= 1)
        m = fmaxf(m, __shfl_xor(m, off, 32));
    if ((tid & 31) == 0) wred[tid >> 5] = m;
    __syncthreads();
    if (tid == 0) {
        float mm = wred[0];
        for (int w = 1; w < NWAVES; ++w) mm = fmaxf(mm, wred[w]);
        rowmax_sh = mm;
    }
    __syncthreads();
    const float M = rowmax_sh;

    // ---- e_i = exp(p_sorted_i - M)
    for (int i = tid; i < ROW_LEN; i += NT)
        sv[i] = expf(__uint_as_float((unsigned int)(kv[i] & 0xFFFFFFFFull)) - M);
    __syncthreads();

    // ---- per-16 suffix sums via WMMA: D = T(16x16, upper-tri ones) x B,
    //      B's 16 columns = 16 independent tiles; one wave handles 256 elements.
    {
        const int wave  = tid >> 5;
        const int lane  = tid & 31;
        const int seg   = wave * 256;          // this wave's 16 tiles
        const int ncol  = lane & 15;           // tile index (B/D column)
        const int kb    = (lane >> 4) * 2;     // K sub-base within a 4-chunk
        const int mrow  = lane & 15;           // A row
        v8f acc = {0.f, 0.f, 0.f, 0.f, 0.f, 0.f, 0.f, 0.f};
#pragma unroll
        for (int c = 0; c < 4; ++c) {          // chain K: 4 x (16x16x4) = 16x16x16
            int k0 = 4 * c + kb;
            v2f a, b;
            a.x = (k0     >= mrow) ? 1.0f : 0.0f;   // T[m][k] = (k >= m)
            a.y = (k0 + 1 >= mrow) ? 1.0f : 0.0f;
            b.x = sv[seg + ncol * 16 + k0];
            b.y = sv[seg + ncol * 16 + k0 + 1];
            acc = __builtin_amdgcn_wmma_f32_16x16x4_f32(
                false, a, false, b, (short)0, acc, false, false);
        }
        // D layout: lane holds column ncol; rows M = r + 8*(lane>>4)
        const int mhalf = (lane >> 4) * 8;
#pragma unroll
        for (int r = 0; r < 8; ++r)
            sv[seg + ncol * 16 + mhalf + r] = acc[r];   // in-tile suffix sums
    }
    __syncthreads();

    // ---- suffix scan of the 256 tile totals (tile total = in-tile suffix at M=0)
    if (tid < 256) tt[tid] = sv[tid * 16];
    __syncthreads();
    for (int off = 1; off < 256; off <<= 1) {
        float add = 0.0f;
        if (tid < 256 && tid + off < 256) add = tt[tid + off];
        __syncthreads();
        if (tid < 256) tt[tid] += add;
        __syncthreads();
    }

    // ---- loss = sum_i ( M + log(S_i) - p_i )
    float loss = 0.0f;
    for (int i = tid; i < ROW_LEN; i += NT) {
        int t = i >> 4;
        float excl = (t < 255) ? tt[t + 1] : 0.0f;      // tiles strictly after t
        float S = sv[i] + excl;
        float pi = __uint_as_float((unsigned int)(kv[i] & 0xFFFFFFFFull));
        loss += logf(S) + M - pi;
    }
    for (int off = 16; off > 0; off >>= 1)
        loss += __shfl_xor(loss, off, 32);
    if ((tid & 31) == 0) wred[tid >> 5] = loss;
    __syncthreads();
    if (tid == 0) {
        float s = 0.0f;
        for (int w = 0; w < NWAVES; ++w) s += wred[w];
        row_loss[row] = s;
    }
}

__global__ __launch_bounds__(256)
void listmle_reduce_kernel(const float* __restrict__ part,
                           float* __restrict__ out, int n, float inv_b)
{
    __shared__ float sh[8];
    float s = 0.0f;
    for (int i = threadIdx.x; i < n; i += 256) s += part[i];
    for (int off = 16; off > 0; off >>= 1) s += __shfl_xor(s, off, 32);
    if ((threadIdx.x & 31) == 0) sh[threadIdx.x >> 5] = s;
    __syncthreads();
    if (threadIdx.x == 0) {
        float t = 0.0f;
        for (int w = 0; w < 8; ++w) t += sh[w];
        out[0] = t * inv_b;
    }
}

extern "C" void kernel_launch(void* const* d_in, const int* in_sizes, int n_in,
                              void* d_out, int out_size, void* d_ws, size_t ws_size,
                              hipStream_t stream) {
    const float* preds  = (const float*)d_in[0];
    const float* labels = (const float*)d_in[1];
    const int B = in_sizes[0] / ROW_LEN;          // 2048
    float* part = (float*)d_ws;                   // B floats of scratch

    listmle_row_kernel<<<B, NT, 0, stream>>>(preds, labels, part);
    listmle_reduce_kernel<<<1, 256, 0, stream>>>(part, (float*)d_out, B,
                                                 1.0f / (float)B);
}